// HANForGraphClassification_60842506715266
// MI455X (gfx1250) — compile-verified
//
#include <hip/hip_runtime.h>
#include <hip/hip_bf16.h>

#define HID 256
#define NHEAD 8
#define NEG_SLOPE 0.2f
#define EPS_F 1e-16f
#define NTILE 4  // 16x64 strip per wave: 4 column tiles share one A fragment

typedef float v2f __attribute__((ext_vector_type(2)));
typedef float v8f __attribute__((ext_vector_type(8)));

// ---------------------------------------------------------------------------
// Core: one wave32 computes a 16x64 fp32 strip of A(NxK) @ B(Kx256) using
// V_WMMA_F32_16X16X4_F32, K-step 4. One A fragment feeds 4 WMMAs.
// A layout (ISA 7.12.2, 32-bit A 16x4): lanes 0-15 -> M=lane, K=k,k+1;
// lanes 16-31 -> M=lane-16, K=k+2,k+3  (one aligned 8-byte load per lane).
// B layout: VGPR0 = rows k / k+2 striped across lane halves, VGPR1 = k+1/k+3.
// C/D: 8 VGPRs, lane<16 -> M=r, lane>=16 -> M=r+8, N=lane&15.
// ---------------------------------------------------------------------------
__device__ __forceinline__ void wmma_row_strip(const float* __restrict__ A,
                                               const float* __restrict__ B,
                                               int K, int row_base, int col_base,
                                               v8f acc[NTILE]) {
  const int lane = threadIdx.x & 31;
  const int half = lane >> 4;
  const int l16  = lane & 15;
  const float* arowp = A + (size_t)(row_base + l16) * (size_t)K;
#pragma unroll
  for (int t = 0; t < NTILE; ++t)
    acc[t] = (v8f){0.f, 0.f, 0.f, 0.f, 0.f, 0.f, 0.f, 0.f};
  for (int k = 0; k < K; k += 4) {
    const int ka = k + half * 2;
    const v2f a = *(const v2f*)(arowp + ka);  // K=ka, ka+1 (8B aligned)
    const float* b0 = B + (size_t)ka * HID + col_base + l16;
#pragma unroll
    for (int t = 0; t < NTILE; ++t) {
      v2f b;
      b.x = b0[t * 16];        // row ka   (or ka+2 for upper half-wave)
      b.y = b0[t * 16 + HID];  // row ka+1 (or ka+3)
      acc[t] = __builtin_amdgcn_wmma_f32_16x16x4_f32(false, a, false, b,
                                                     (short)0, acc[t], false, false);
    }
  }
}

// C = A @ B + bias  (B is Kx256 row-major, C is Nx256); 16x64 strip per wave
__global__ void gemm_bias_kernel(const float* __restrict__ A,
                                 const float* __restrict__ B,
                                 const float* __restrict__ bias,
                                 float* __restrict__ C, int K) {
  const int row_base = blockIdx.x * 16;
  const int col_base = blockIdx.y * (16 * NTILE);
  v8f acc[NTILE];
  wmma_row_strip(A, B, K, row_base, col_base, acc);
  const int lane = threadIdx.x & 31;
  const int half = lane >> 4;
  const int l16  = lane & 15;
  const int crow = row_base + half * 8;
#pragma unroll
  for (int t = 0; t < NTILE; ++t) {
    const int col = col_base + t * 16 + l16;
    const float bv = bias[col];
#pragma unroll
    for (int r = 0; r < 8; ++r)
      C[(size_t)(crow + r) * HID + col] = acc[t][r] + bv;
  }
}

// score += sum over strip of tanh((X@Wk)[n,c] + bk[c]) * q[c]
__global__ void sem_score_kernel(const float* __restrict__ X,
                                 const float* __restrict__ Wk,
                                 const float* __restrict__ bk,
                                 const float* __restrict__ q,
                                 float* __restrict__ score) {
  const int row_base = blockIdx.x * 16;
  const int col_base = blockIdx.y * (16 * NTILE);
  v8f acc[NTILE];
  wmma_row_strip(X, Wk, HID, row_base, col_base, acc);
  const int lane = threadIdx.x & 31;
  const int l16  = lane & 15;
  float p = 0.f;
#pragma unroll
  for (int t = 0; t < NTILE; ++t) {
    const int col = col_base + t * 16 + l16;
    const float bv = bk[col], qv = q[col];
#pragma unroll
    for (int r = 0; r < 8; ++r) p += tanhf(acc[t][r] + bv) * qv;
  }
#pragma unroll
  for (int off = 16; off > 0; off >>= 1) p += __shfl_down(p, off, 32);
  if (lane == 0) atomicAdd(score, p);
}

// ---------------------------------------------------------------------------
// Graph-attention (bandwidth-bound) kernels
// ---------------------------------------------------------------------------
__global__ void fill_kernel(float* __restrict__ p, float v, int n) {
  int i = blockIdx.x * blockDim.x + threadIdx.x;
  if (i < n) p[i] = v;
}

// alpha[n,h] = sum_d h[n, h*32+d] * att[h*32+d]
__global__ void node_alpha_kernel(const float* __restrict__ h,
                                  const float* __restrict__ att,
                                  float* __restrict__ alpha, int n_nodes) {
  int tid = blockIdx.x * blockDim.x + threadIdx.x;
  if (tid >= n_nodes * NHEAD) return;
  const int n = tid >> 3, hh = tid & 7;
  const float* hp = h + (size_t)n * HID + hh * 32;
  const float* ap = att + hh * 32;
  float s = 0.f;
#pragma unroll
  for (int d = 0; d < 32; ++d) s += hp[d] * ap[d];
  alpha[tid] = s;
}

__device__ __forceinline__ float leaky(float a) {
  return a > 0.f ? a : NEG_SLOPE * a;
}

__device__ __forceinline__ void atomicMaxFloat(float* addr, float val) {
  unsigned int* ua = reinterpret_cast<unsigned int*>(addr);
  unsigned int old = __float_as_uint(*addr);
  while (__uint_as_float(old) < val) {
    unsigned int assumed = old;
    old = atomicCAS(ua, assumed, __float_as_uint(val));
    if (old == assumed) break;
  }
}

__global__ void edge_amax_kernel(const int* __restrict__ src, const int* __restrict__ dst,
                                 const float* __restrict__ as_, const float* __restrict__ ad_,
                                 float* __restrict__ amax, int E) {
  int tid = blockIdx.x * blockDim.x + threadIdx.x;
  if (tid >= E * NHEAD) return;
  const int e = tid >> 3, hh = tid & 7;
  const int s = src[e], d = dst[e];
  const float a = leaky(as_[s * NHEAD + hh] + ad_[d * NHEAD + hh]);
  atomicMaxFloat(&amax[d * NHEAD + hh], a);
}

__global__ void edge_denom_kernel(const int* __restrict__ src, const int* __restrict__ dst,
                                  const float* __restrict__ as_, const float* __restrict__ ad_,
                                  const float* __restrict__ amax, float* __restrict__ denom, int E) {
  int tid = blockIdx.x * blockDim.x + threadIdx.x;
  if (tid >= E * NHEAD) return;
  const int e = tid >> 3, hh = tid & 7;
  const int s = src[e], d = dst[e];
  const float a = leaky(as_[s * NHEAD + hh] + ad_[d * NHEAD + hh]);
  atomicAdd(&denom[d * NHEAD + hh], expf(a - amax[d * NHEAD + hh]));
}

__global__ void edge_attn_kernel(const int* __restrict__ src, const int* __restrict__ dst,
                                 const float* __restrict__ as_, const float* __restrict__ ad_,
                                 const float* __restrict__ amax, const float* __restrict__ denom,
                                 float* __restrict__ attn, int E) {
  int tid = blockIdx.x * blockDim.x + threadIdx.x;
  if (tid >= E * NHEAD) return;
  const int e = tid >> 3, hh = tid & 7;
  const int s = src[e], d = dst[e];
  const float a = leaky(as_[s * NHEAD + hh] + ad_[d * NHEAD + hh]);
  const float ex = expf(a - amax[d * NHEAD + hh]);
  attn[tid] = ex / (denom[d * NHEAD + hh] + EPS_F);
}

// out[dst, c] += xs[src, c] * attn[e, c/32]; one block (256 thr) per edge
__global__ void edge_scatter_kernel(const int* __restrict__ src, const int* __restrict__ dst,
                                    const float* __restrict__ xs,
                                    const float* __restrict__ attn,
                                    float* __restrict__ out) {
  const int e = blockIdx.x;
  const int c = threadIdx.x;
  const int s = src[e], d = dst[e];
  const float w = attn[e * NHEAD + (c >> 5)];
  atomicAdd(&out[(size_t)d * HID + c], xs[(size_t)s * HID + c] * w);
}

__global__ void relu_kernel(float* __restrict__ p, int n) {
  int i = blockIdx.x * blockDim.x + threadIdx.x;
  if (i < n) p[i] = fmaxf(p[i], 0.f);
}

// column sums: thread c owns column c over a row chunk, one atomic per chunk
#define CS_ROWS 512
__global__ void colsum_kernel(const float* __restrict__ X, float* __restrict__ sums, int n_rows) {
  const int c = threadIdx.x;
  int r0 = blockIdx.x * CS_ROWS;
  int r1 = r0 + CS_ROWS; if (r1 > n_rows) r1 = n_rows;
  float s = 0.f;
  for (int r = r0; r < r1; ++r) s += X[(size_t)r * HID + c];
  atomicAdd(&sums[c], s);
}

// Final head: semantic softmax (2 metapaths for user, 1 for post == weight 1.0),
// mean concat -> 512x256 lin -> relu -> 256x2 classifier.
__global__ void head_kernel(const float* __restrict__ sum_uu, const float* __restrict__ sum_pu,
                            const float* __restrict__ sum_up, const float* __restrict__ scores,
                            const float* __restrict__ W_lin, const float* __restrict__ b_lin,
                            const float* __restrict__ W_cls, const float* __restrict__ b_cls,
                            float* __restrict__ out, int n_user, int n_post) {
  __shared__ float g[2 * HID];
  __shared__ float hbuf[HID];
  const int t = threadIdx.x;  // 256 threads
  const float inv_u = 1.f / (float)n_user;
  const float s0 = scores[0] * inv_u;
  const float s1 = scores[1] * inv_u;
  const float m  = fmaxf(s0, s1);
  const float e0 = expf(s0 - m), e1 = expf(s1 - m);
  const float w0 = e0 / (e0 + e1), w1 = e1 / (e0 + e1);
  g[t]       = (w0 * sum_uu[t] + w1 * sum_pu[t]) * inv_u;
  g[HID + t] = sum_up[t] / (float)n_post;
  __syncthreads();
  float acc = b_lin[t];
  for (int i = 0; i < 2 * HID; ++i) acc += g[i] * W_lin[i * HID + t];
  hbuf[t] = fmaxf(acc, 0.f);
  __syncthreads();
  if (t < 2) {
    float o = b_cls[t];
    for (int j = 0; j < HID; ++j) o += hbuf[j] * W_cls[j * 2 + t];
    out[t] = o;
  }
}

// ---------------------------------------------------------------------------
// Host-side pipeline
// ---------------------------------------------------------------------------
static void run_edge_type(const float* h_src, int n_src, const float* h_dst, int n_dst,
                          const float* att_s, const float* att_d,
                          const int* src, const int* dst, int E,
                          float* as_buf, float* ad_buf, float* amax, float* denom,
                          float* attn, float* out, hipStream_t stream) {
  const int T = 256;
  fill_kernel<<<(n_dst * NHEAD + T - 1) / T, T, 0, stream>>>(amax, -__builtin_inff(), n_dst * NHEAD);
  fill_kernel<<<(n_dst * NHEAD + T - 1) / T, T, 0, stream>>>(denom, 0.f, n_dst * NHEAD);
  fill_kernel<<<(n_dst * HID + T - 1) / T, T, 0, stream>>>(out, 0.f, n_dst * HID);
  node_alpha_kernel<<<(n_src * NHEAD + T - 1) / T, T, 0, stream>>>(h_src, att_s, as_buf, n_src);
  node_alpha_kernel<<<(n_dst * NHEAD + T - 1) / T, T, 0, stream>>>(h_dst, att_d, ad_buf, n_dst);
  const int EH = E * NHEAD;
  edge_amax_kernel <<<(EH + T - 1) / T, T, 0, stream>>>(src, dst, as_buf, ad_buf, amax, E);
  edge_denom_kernel<<<(EH + T - 1) / T, T, 0, stream>>>(src, dst, as_buf, ad_buf, amax, denom, E);
  edge_attn_kernel <<<(EH + T - 1) / T, T, 0, stream>>>(src, dst, as_buf, ad_buf, amax, denom, attn, E);
  edge_scatter_kernel<<<E, HID, 0, stream>>>(src, dst, h_src, attn, out);
  relu_kernel<<<(n_dst * HID + T - 1) / T, T, 0, stream>>>(out, n_dst * HID);
}

extern "C" void kernel_launch(void* const* d_in, const int* in_sizes, int n_in,
                              void* d_out, int out_size, void* d_ws, size_t ws_size,
                              hipStream_t stream) {
  const float* x_user  = (const float*)d_in[0];
  const float* x_post  = (const float*)d_in[1];
  const int*   e_uu    = (const int*)d_in[2];
  const int*   e_up    = (const int*)d_in[3];
  const int*   e_pu    = (const int*)d_in[4];
  const float* W_user  = (const float*)d_in[5];
  const float* b_user  = (const float*)d_in[6];
  const float* W_post  = (const float*)d_in[7];
  const float* b_post  = (const float*)d_in[8];
  const float* att_s_uu = (const float*)d_in[9];
  const float* att_d_uu = (const float*)d_in[10];
  const float* att_s_up = (const float*)d_in[11];
  const float* att_d_up = (const float*)d_in[12];
  const float* att_s_pu = (const float*)d_in[13];
  const float* att_d_pu = (const float*)d_in[14];
  const float* q_sem   = (const float*)d_in[15];
  const float* Wk_sem  = (const float*)d_in[16];
  const float* bk_sem  = (const float*)d_in[17];
  const float* W_lin   = (const float*)d_in[18];
  const float* b_lin   = (const float*)d_in[19];
  const float* W_cls   = (const float*)d_in[20];
  const float* b_cls   = (const float*)d_in[21];

  const int IN_U = 256, IN_P = 128;
  const int n_user = in_sizes[0] / IN_U;   // 100000 (divisible by 16)
  const int n_post = in_sizes[1] / IN_P;   // 200000 (divisible by 16)
  const int E      = in_sizes[2] / 2;      // 500000
  const int n_max  = (n_user > n_post) ? n_user : n_post;

  // workspace carve-out
  char* ws = (char*)d_ws;
  size_t off = 0;
  auto carve = [&](size_t bytes) -> float* {
    float* p = (float*)(ws + off);
    off += (bytes + 255) & ~(size_t)255;
    return p;
  };
  float* hu     = carve((size_t)n_user * HID * 4);
  float* hp     = carve((size_t)n_post * HID * 4);
  float* out_uu = carve((size_t)n_user * HID * 4);
  float* out_pu = carve((size_t)n_user * HID * 4);
  float* out_up = carve((size_t)n_post * HID * 4);
  float* as_buf = carve((size_t)n_max * NHEAD * 4);
  float* ad_buf = carve((size_t)n_max * NHEAD * 4);
  float* amax   = carve((size_t)n_max * NHEAD * 4);
  float* denom  = carve((size_t)n_max * NHEAD * 4);
  float* attn   = carve((size_t)E * NHEAD * 4);
  float* sum_uu = carve(HID * 4);
  float* sum_pu = carve(HID * 4);
  float* sum_up = carve(HID * 4);
  float* scores = carve(2 * 4);
  (void)ws_size; (void)n_in; (void)out_size;

  const int colblk = HID / (16 * NTILE);  // 4 column strips of 64

  // 1) per-node-type projections (WMMA f32 GEMMs, 16x64 strip per wave)
  gemm_bias_kernel<<<dim3(n_user / 16, colblk), 32, 0, stream>>>(x_user, W_user, b_user, hu, IN_U);
  gemm_bias_kernel<<<dim3(n_post / 16, colblk), 32, 0, stream>>>(x_post, W_post, b_post, hp, IN_P);

  // 2) node-level attention per edge type
  run_edge_type(hu, n_user, hu, n_user, att_s_uu, att_d_uu, e_uu, e_uu + E, E,
                as_buf, ad_buf, amax, denom, attn, out_uu, stream);
  run_edge_type(hu, n_user, hp, n_post, att_s_up, att_d_up, e_up, e_up + E, E,
                as_buf, ad_buf, amax, denom, attn, out_up, stream);
  run_edge_type(hp, n_post, hu, n_user, att_s_pu, att_d_pu, e_pu, e_pu + E, E,
                as_buf, ad_buf, amax, denom, attn, out_pu, stream);

  // 3) semantic scores for the user metapaths (post has a single metapath ->
  //    softmax weight is exactly 1.0, its score GEMM is mathematically dead)
  fill_kernel<<<1, 32, 0, stream>>>(scores, 0.f, 2);
  sem_score_kernel<<<dim3(n_user / 16, colblk), 32, 0, stream>>>(out_uu, Wk_sem, bk_sem, q_sem, scores + 0);
  sem_score_kernel<<<dim3(n_user / 16, colblk), 32, 0, stream>>>(out_pu, Wk_sem, bk_sem, q_sem, scores + 1);

  // 4) column sums for mean pooling
  fill_kernel<<<1, 256, 0, stream>>>(sum_uu, 0.f, HID);
  fill_kernel<<<1, 256, 0, stream>>>(sum_pu, 0.f, HID);
  fill_kernel<<<1, 256, 0, stream>>>(sum_up, 0.f, HID);
  colsum_kernel<<<(n_user + CS_ROWS - 1) / CS_ROWS, HID, 0, stream>>>(out_uu, sum_uu, n_user);
  colsum_kernel<<<(n_user + CS_ROWS - 1) / CS_ROWS, HID, 0, stream>>>(out_pu, sum_pu, n_user);
  colsum_kernel<<<(n_post + CS_ROWS - 1) / CS_ROWS, HID, 0, stream>>>(out_up, sum_up, n_post);

  // 5) semantic softmax + MLP head
  head_kernel<<<1, HID, 0, stream>>>(sum_uu, sum_pu, sum_up, scores,
                                     W_lin, b_lin, W_cls, b_cls,
                                     (float*)d_out, n_user, n_post);
}